// GCN_85676007621063
// MI455X (gfx1250) — compile-verified
//
#include <hip/hip_runtime.h>
#include <hip/hip_bf16.h>

typedef __attribute__((ext_vector_type(2))) float v2f;
typedef __attribute__((ext_vector_type(8))) float v8f;

#define NN 50000
#define EE 800000
#define DD 128

// ---------------------------------------------------------------------------
// Degree accumulation: deg[dst] += 1 per edge (self-loop +1 folded into dinv).
// ---------------------------------------------------------------------------
__global__ __launch_bounds__(256) void deg_kernel(const int* __restrict__ ei,
                                                  float* __restrict__ deg) {
    int e = blockIdx.x * blockDim.x + threadIdx.x;
    if (e < EE) {
        unsafeAtomicAdd(&deg[ei[EE + e]], 1.0f);
    }
}

__global__ __launch_bounds__(256) void dinv_kernel(const float* __restrict__ deg,
                                                   float* __restrict__ dinv) {
    int i = blockIdx.x * blockDim.x + threadIdx.x;
    if (i < NN) {
        dinv[i] = rsqrtf(deg[i] + 1.0f);   // +1 = self loop; always > 0
    }
}

// ---------------------------------------------------------------------------
// T[N,128] = X[N,128] @ W[128,128] in fp32 via V_WMMA_F32_16X16X4_F32.
// Block = 256 threads = 8 waves; wave w computes the 16x16 tile at column
// tile w; block b handles rows [16b, 16b+16). 50000 % 16 == 0.
//
// 16x4 f32 A layout : lane m=lane&15; VGPR0 = A[m][k0+2*half], VGPR1 = +1
// 4x16 f32 B layout : lane n=lane&15; VGPR0 = B[k0+2*half][n], VGPR1 = +1
// 16x16 f32 C layout: VGPR v -> row (v + 8*half), col = lane&15
// ---------------------------------------------------------------------------
__global__ __launch_bounds__(256) void gemm_wmma(const float* __restrict__ X,
                                                 const float* __restrict__ W,
                                                 float* __restrict__ T) {
    const int wave = threadIdx.x >> 5;     // 0..7 column tile
    const int lane = threadIdx.x & 31;
    const int half = lane >> 4;            // 0 or 1
    const int mn   = lane & 15;            // row of A frag / col of B frag
    const int m0   = blockIdx.x * 16;
    const int n0   = wave * 16;

    const float* xrow = X + (size_t)(m0 + mn) * DD;
    const float* wcol = W + n0 + mn;

    v8f c = {};
    for (int kk = 0; kk < DD; kk += 4) {
        const int k0 = kk + 2 * half;
        v2f a, b;
        a.x = xrow[k0];
        a.y = xrow[k0 + 1];
        b.x = wcol[(size_t)k0 * DD];
        b.y = wcol[(size_t)(k0 + 1) * DD];
        c = __builtin_amdgcn_wmma_f32_16x16x4_f32(
                /*neg_a=*/false, a, /*neg_b=*/false, b,
                /*c_mod=*/(short)0, c, /*reuse_a=*/false, /*reuse_b=*/false);
    }

    float* trow = T + (size_t)(m0 + 8 * half) * DD + n0 + mn;
    #pragma unroll
    for (int v = 0; v < 8; ++v) {
        trow[(size_t)v * DD] = c[v];
    }
}

// ---------------------------------------------------------------------------
// Edge scatter: agg[dst] += dinv[src]*dinv[dst] * T[src].
// One wave per edge; each lane owns 4 channels (float4 load, 4 f32 atomics).
// ---------------------------------------------------------------------------
__global__ __launch_bounds__(256) void scatter_kernel(const int* __restrict__ ei,
                                                      const float* __restrict__ t,
                                                      const float* __restrict__ dinv,
                                                      float* __restrict__ agg) {
    const int gid  = blockIdx.x * blockDim.x + threadIdx.x;
    const int lane = gid & 31;
    const int wid  = gid >> 5;
    const int nwav = (gridDim.x * blockDim.x) >> 5;

    for (int e = wid; e < EE; e += nwav) {
        const int s = ei[e];
        const int d = ei[EE + e];
        const float norm = dinv[s] * dinv[d];
        const float4 v = *reinterpret_cast<const float4*>(t + (size_t)s * DD + lane * 4);
        float* o = agg + (size_t)d * DD + lane * 4;
        unsafeAtomicAdd(o + 0, norm * v.x);
        unsafeAtomicAdd(o + 1, norm * v.y);
        unsafeAtomicAdd(o + 2, norm * v.z);
        unsafeAtomicAdd(o + 3, norm * v.w);
    }
}

// ---------------------------------------------------------------------------
// h = relu(agg + dinv[i]^2 * t[i] + b)   (self-loop + bias + activation)
// ---------------------------------------------------------------------------
__global__ __launch_bounds__(256) void finalize_kernel(const float* __restrict__ agg,
                                                       const float* __restrict__ t,
                                                       const float* __restrict__ dinv,
                                                       const float* __restrict__ b,
                                                       float* __restrict__ h) {
    int idx = blockIdx.x * blockDim.x + threadIdx.x;
    if (idx < NN * DD) {
        const int row = idx >> 7;
        const float di = dinv[row];
        float v = agg[idx] + di * di * t[idx] + b[idx & (DD - 1)];
        h[idx] = v > 0.0f ? v : 0.0f;
    }
}

// ---------------------------------------------------------------------------
// out[i] = dot(h[i,:], Wr) + br   — one wave per node, shfl_xor reduction.
// ---------------------------------------------------------------------------
__global__ __launch_bounds__(256) void readout_kernel(const float* __restrict__ h,
                                                      const float* __restrict__ Wr,
                                                      const float* __restrict__ br,
                                                      float* __restrict__ out) {
    const int gid  = blockIdx.x * blockDim.x + threadIdx.x;
    const int lane = gid & 31;
    const int i    = gid >> 5;
    if (i >= NN) return;

    const float4 v = *reinterpret_cast<const float4*>(h + (size_t)i * DD + lane * 4);
    const float4 w = *reinterpret_cast<const float4*>(Wr + lane * 4);
    float s = v.x * w.x + v.y * w.y + v.z * w.z + v.w * w.w;
    s += __shfl_xor(s, 16, 32);
    s += __shfl_xor(s, 8, 32);
    s += __shfl_xor(s, 4, 32);
    s += __shfl_xor(s, 2, 32);
    s += __shfl_xor(s, 1, 32);
    if (lane == 0) out[i] = s + br[0];
}

// ---------------------------------------------------------------------------
extern "C" void kernel_launch(void* const* d_in, const int* in_sizes, int n_in,
                              void* d_out, int out_size, void* d_ws, size_t ws_size,
                              hipStream_t stream) {
    const float* x  = (const float*)d_in[0];
    const int*   ei = (const int*)d_in[1];
    const float* W1 = (const float*)d_in[2];
    const float* b1 = (const float*)d_in[3];
    const float* W2 = (const float*)d_in[4];
    const float* b2 = (const float*)d_in[5];
    const float* Wr = (const float*)d_in[6];
    const float* br = (const float*)d_in[7];
    float* out = (float*)d_out;

    // Workspace layout (floats): t | agg | h | deg | dinv
    float* t    = (float*)d_ws;
    float* agg  = t   + (size_t)NN * DD;
    float* h    = agg + (size_t)NN * DD;
    float* deg  = h   + (size_t)NN * DD;
    float* dinv = deg + NN;

    const size_t featBytes = (size_t)NN * DD * sizeof(float);

    // Degrees + symmetric norm factors
    hipMemsetAsync(deg, 0, (size_t)NN * sizeof(float), stream);
    deg_kernel<<<(EE + 255) / 256, 256, 0, stream>>>(ei, deg);
    dinv_kernel<<<(NN + 255) / 256, 256, 0, stream>>>(deg, dinv);

    // ---- Layer 1 ----
    gemm_wmma<<<NN / 16, 256, 0, stream>>>(x, W1, t);
    hipMemsetAsync(agg, 0, featBytes, stream);
    scatter_kernel<<<8192, 256, 0, stream>>>(ei, t, dinv, agg);
    finalize_kernel<<<(NN * DD + 255) / 256, 256, 0, stream>>>(agg, t, dinv, b1, h);

    // ---- Layer 2 ----
    gemm_wmma<<<NN / 16, 256, 0, stream>>>(h, W2, t);
    hipMemsetAsync(agg, 0, featBytes, stream);
    scatter_kernel<<<8192, 256, 0, stream>>>(ei, t, dinv, agg);
    finalize_kernel<<<(NN * DD + 255) / 256, 256, 0, stream>>>(agg, t, dinv, b2, h);

    // ---- Readout ----
    readout_kernel<<<(NN * 32 + 255) / 256, 256, 0, stream>>>(h, Wr, br, out);
}